// TransformerEncoder_88304527606667
// MI455X (gfx1250) — compile-verified
//
#include <hip/hip_runtime.h>
#include <math.h>

typedef float v2f __attribute__((ext_vector_type(2)));
typedef float v8f __attribute__((ext_vector_type(8)));

#define NODES 50000
#define EDGES 500000
#define DIM   64
#define HEADS 4
#define HD    256   /* HEADS*DIM */
#define NGR   64
#define NLAY  3

// ---------------------------------------------------------------------------
// Y[M,NOUT] = X[M,64] @ W[64,NOUT] + bias   via V_WMMA_F32_16X16X4_F32.
// One wave computes a 16x64 strip: 4 accumulator tiles share one A fragment
// per K-step (16 K-steps * 4 tiles = 64 WMMA issues, fully unrolled).
// NOUT is a template constant so all W/Y addresses fold to immediate offsets.
// A 16x4 layout: lanes 0-15 hold M=0..15 K={0,1}; lanes 16-31 hold K={2,3}.
// B 4x16 layout: VGPR0 = rows K=0/K=2 per half-wave, VGPR1 = K=1/K=3.
// C/D 16x16: VGPR j = row j (lanes 0-15) and row j+8 (lanes 16-31).
// ---------------------------------------------------------------------------
template <int NOUT>
__global__ __launch_bounds__(32)
void wmma_gemm_f32(const float* __restrict__ X, const float* __restrict__ W,
                   const float* __restrict__ bias, float* __restrict__ Y) {
  const int lane = threadIdx.x & 31;
  const int hi   = lane >> 4;            // half-wave select
  const int l15  = lane & 15;
  const int m0   = blockIdx.x * 16;
  const int n0   = blockIdx.y * 64;      // 4 consecutive 16-wide tiles
  const float* Xr = X + (size_t)(m0 + l15) * DIM;
  const float* Wc = W + n0 + l15;        // per-lane column base
  v8f acc[4];
#pragma unroll
  for (int t = 0; t < 4; ++t) acc[t] = v8f{0.f,0.f,0.f,0.f,0.f,0.f,0.f,0.f};
#pragma unroll
  for (int k = 0; k < DIM; k += 4) {
    const int ks = k + (hi << 1);
    v2f a; a.x = Xr[ks]; a.y = Xr[ks + 1];
    const float* Wk0 = Wc + ks * NOUT;   // row ks base; rest are immediates
#pragma unroll
    for (int t = 0; t < 4; ++t) {
      v2f b; b.x = Wk0[t * 16]; b.y = Wk0[NOUT + t * 16];
      acc[t] = __builtin_amdgcn_wmma_f32_16x16x4_f32(false, a, false, b,
                                                     (short)0, acc[t],
                                                     false, false);
    }
  }
#pragma unroll
  for (int t = 0; t < 4; ++t) {
    const int col = n0 + t * 16 + l15;
    const float bv = bias[col];
#pragma unroll
    for (int j = 0; j < 8; ++j) {
      const int row = m0 + j + (hi << 3);
      Y[(size_t)row * NOUT + col] = acc[t][j] + bv;
    }
  }
}

// ---------------------------------------------------------------------------
// Pass A: per-edge attention logits + ordered-uint segment max.
// One wave per edge. lane = head*8 + chunk; each lane loads 8 contiguous f32.
// ---------------------------------------------------------------------------
__device__ __forceinline__ unsigned f32_to_ordkey(float f) {
  unsigned u = __float_as_uint(f);
  return (u & 0x80000000u) ? ~u : (u | 0x80000000u);
}
__device__ __forceinline__ float ordkey_to_f32(unsigned key) {
  unsigned u = (key & 0x80000000u) ? (key & 0x7FFFFFFFu) : ~key;
  return __uint_as_float(u);
}

__global__ __launch_bounds__(256)
void edge_alpha(const float* __restrict__ q, const float* __restrict__ k,
                const int* __restrict__ src, const int* __restrict__ dst,
                float* __restrict__ alpha, unsigned* __restrict__ mkey) {
  const int e = blockIdx.x * (blockDim.x >> 5) + (threadIdx.x >> 5);
  if (e >= EDGES) return;
  const int lane = threadIdx.x & 31;
  const int h = lane >> 3, sub = lane & 7;
  const int s = src[e], t = dst[e];
  const float4* qp = (const float4*)(q + (size_t)t * HD + h * DIM + sub * 8);
  const float4* kp = (const float4*)(k + (size_t)s * HD + h * DIM + sub * 8);
  const float4 q0 = qp[0], q1 = qp[1];
  const float4 k0 = kp[0], k1 = kp[1];
  float dot = q0.x * k0.x + q0.y * k0.y + q0.z * k0.z + q0.w * k0.w
            + q1.x * k1.x + q1.y * k1.y + q1.z * k1.z + q1.w * k1.w;
  dot += __shfl_xor(dot, 1);
  dot += __shfl_xor(dot, 2);
  dot += __shfl_xor(dot, 4);
  dot *= 0.125f;                       // SCALE = 1/sqrt(64)
  if (sub == 0) {
    alpha[e * HEADS + h] = dot;
    atomicMax(&mkey[t * HEADS + h], f32_to_ordkey(dot));
  }
}

// ---------------------------------------------------------------------------
// Pass B: e = exp(alpha - m[dst]); den[dst] += e.  One thread per (edge,head).
// ---------------------------------------------------------------------------
__global__ __launch_bounds__(256)
void edge_exp(const int* __restrict__ dst, float* __restrict__ alpha,
              const unsigned* __restrict__ mkey, float* __restrict__ den) {
  const int i = blockIdx.x * blockDim.x + threadIdx.x;
  if (i >= EDGES * HEADS) return;
  const int e = i >> 2, h = i & 3;
  const int t = dst[e];
  const float m = ordkey_to_f32(mkey[t * HEADS + h]);
  const float ev = expf(alpha[i] - m);
  alpha[i] = ev;
  atomicAdd(&den[t * HEADS + h], ev);
}

// ---------------------------------------------------------------------------
// Pass C: agg[dst,d] += (1/H) * sum_h a[e,h] * v[src,h,d].
// One wave per edge; cross-head fold via shfl_xor(8/16) -> 64 atomics/edge.
// ---------------------------------------------------------------------------
__global__ __launch_bounds__(256)
void edge_scatter(const float* __restrict__ v, const int* __restrict__ src,
                  const int* __restrict__ dst, const float* __restrict__ alpha,
                  const float* __restrict__ den, float* __restrict__ agg) {
  const int e = blockIdx.x * (blockDim.x >> 5) + (threadIdx.x >> 5);
  if (e >= EDGES) return;
  const int lane = threadIdx.x & 31;
  const int h = lane >> 3, sub = lane & 7;
  const int s = src[e], t = dst[e];
  const float a = alpha[e * HEADS + h] / (den[t * HEADS + h] + 1e-16f);
  const float w = a * 0.25f;           // head mean (concat=False)
  const float4* vp = (const float4*)(v + (size_t)s * HD + h * DIM + sub * 8);
  const float4 v0 = vp[0], v1 = vp[1];
  float r[8] = {v0.x * w, v0.y * w, v0.z * w, v0.w * w,
                v1.x * w, v1.y * w, v1.z * w, v1.w * w};
#pragma unroll
  for (int j = 0; j < 8; ++j) {
    float t0 = r[j];
    t0 += __shfl_xor(t0, 8);
    t0 += __shfl_xor(t0, 16);          // sum over the 4 heads
    r[j] = t0;
  }
  if (h == 0) {
    float* dp = agg + (size_t)t * DIM + sub * 8;
#pragma unroll
    for (int j = 0; j < 8; ++j) atomicAdd(&dp[j], r[j]);
  }
}

// ---------------------------------------------------------------------------
// Combine: hout = agg + skip, optional ReLU. (agg may alias hout.)
// ---------------------------------------------------------------------------
__global__ __launch_bounds__(256)
void combine(const float* __restrict__ agg, const float* __restrict__ skip,
             float* __restrict__ hout, int relu) {
  const int i = blockIdx.x * blockDim.x + threadIdx.x;
  if (i >= NODES * DIM) return;
  float val = agg[i] + skip[i];
  if (relu) val = fmaxf(val, 0.f);
  hout[i] = val;
}

// ---------------------------------------------------------------------------
// Graph pooling: segment mean over batch.
// ---------------------------------------------------------------------------
__global__ __launch_bounds__(256)
void pool_acc(const float* __restrict__ h, const int* __restrict__ batch,
              float* __restrict__ gsum, float* __restrict__ gcnt) {
  const int i = blockIdx.x * blockDim.x + threadIdx.x;
  if (i >= NODES * DIM) return;
  const int n = i >> 6, d = i & 63;
  const int g = batch[n];
  atomicAdd(&gsum[g * DIM + d], h[i]);
  if (d == 0) atomicAdd(&gcnt[g], 1.0f);
}

__global__ __launch_bounds__(256)
void pool_final(const float* __restrict__ gsum, const float* __restrict__ gcnt,
                float* __restrict__ grepr) {
  const int i = blockIdx.x * blockDim.x + threadIdx.x;
  if (i >= NGR * DIM) return;
  grepr[i] = gsum[i] / fmaxf(gcnt[i >> 6], 1.0f);
}

// ---------------------------------------------------------------------------
extern "C" void kernel_launch(void* const* d_in, const int* in_sizes, int n_in,
                              void* d_out, int out_size, void* d_ws,
                              size_t ws_size, hipStream_t stream) {
  const float* x     = (const float*)d_in[0];
  const int*   ei    = (const int*)d_in[1];    // [2,E] row-major
  const int*   batch = (const int*)d_in[2];
  const int* srcIdx = ei;
  const int* dstIdx = ei + EDGES;

  float* ws = (float*)d_ws;
  size_t off = 0;
  float*    q     = ws + off; off += (size_t)NODES * HD;
  float*    kbuf  = ws + off; off += (size_t)NODES * HD;
  float*    vbuf  = ws + off; off += (size_t)NODES * HD;
  float*    skip  = ws + off; off += (size_t)NODES * DIM;
  float*    alpha = ws + off; off += (size_t)EDGES * HEADS;
  unsigned* mkey  = (unsigned*)(ws + off); off += (size_t)NODES * HEADS;
  float*    den   = ws + off; off += (size_t)NODES * HEADS;
  float*    aggA  = ws + off; off += (size_t)NODES * DIM;
  float*    aggB  = ws + off; off += (size_t)NODES * DIM;
  float*    gsum  = ws + off; off += (size_t)NGR * DIM;
  float*    gcnt  = ws + off; off += (size_t)NGR;

  float* nodeOut  = (float*)d_out;                      // [NODES*DIM]
  float* graphOut = nodeOut + (size_t)NODES * DIM;      // [NGR*DIM]

  const dim3 gQKV(NODES / 16, HD / 64);   // 3125 x 4 strips, exact
  const dim3 gSkp(NODES / 16, DIM / 64);  // 3125 x 1 strips, exact
  const int edgeBlocks = (EDGES + 7) / 8; // 8 waves (edges) per 256-thr block
  const int ehBlocks   = (EDGES * HEADS + 255) / 256;
  const int ndBlocks   = (NODES * DIM + 255) / 256;

  const float* hin = x;
  for (int layer = 0; layer < NLAY; ++layer) {
    const float* Wq = (const float*)d_in[3 + layer * 8 + 0];
    const float* bq = (const float*)d_in[3 + layer * 8 + 1];
    const float* Wk = (const float*)d_in[3 + layer * 8 + 2];
    const float* bk = (const float*)d_in[3 + layer * 8 + 3];
    const float* Wv = (const float*)d_in[3 + layer * 8 + 4];
    const float* bv = (const float*)d_in[3 + layer * 8 + 5];
    const float* Wsk= (const float*)d_in[3 + layer * 8 + 6];
    const float* bsk= (const float*)d_in[3 + layer * 8 + 7];
    float* agg = (layer & 1) ? aggB : aggA;

    wmma_gemm_f32<HD> <<<gQKV, 32, 0, stream>>>(hin, Wq, bq, q);
    wmma_gemm_f32<HD> <<<gQKV, 32, 0, stream>>>(hin, Wk, bk, kbuf);
    wmma_gemm_f32<HD> <<<gQKV, 32, 0, stream>>>(hin, Wv, bv, vbuf);
    wmma_gemm_f32<DIM><<<gSkp, 32, 0, stream>>>(hin, Wsk, bsk, skip);

    hipMemsetAsync(mkey, 0, sizeof(unsigned) * NODES * HEADS, stream);
    hipMemsetAsync(den,  0, sizeof(float) * NODES * HEADS, stream);
    hipMemsetAsync(agg,  0, sizeof(float) * NODES * DIM, stream);

    edge_alpha  <<<edgeBlocks, 256, 0, stream>>>(q, kbuf, srcIdx, dstIdx,
                                                 alpha, mkey);
    edge_exp    <<<ehBlocks,   256, 0, stream>>>(dstIdx, alpha, mkey, den);
    edge_scatter<<<edgeBlocks, 256, 0, stream>>>(vbuf, srcIdx, dstIdx, alpha,
                                                 den, agg);

    float* hout = (layer == NLAY - 1) ? nodeOut : agg;  // in-place ping-pong
    combine<<<ndBlocks, 256, 0, stream>>>(agg, skip, hout, layer != NLAY - 1);
    hin = hout;
  }

  hipMemsetAsync(gsum, 0, sizeof(float) * (NGR * DIM + NGR), stream); // +gcnt
  pool_acc  <<<ndBlocks, 256, 0, stream>>>(nodeOut, batch, gsum, gcnt);
  pool_final<<<(NGR * DIM + 255) / 256, 256, 0, stream>>>(gsum, gcnt, graphOut);
}